// CapsuleLikelihood_58523224375784
// MI455X (gfx1250) — compile-verified
//
#include <hip/hip_runtime.h>
#include <hip/hip_bf16.h>
#include <stdint.h>

// Toggle async-DMA staging (gfx1250 GLOBAL_LOAD_ASYNC_TO_LDS_B128).
#ifndef USE_ASYNC_LDS
#define USE_ASYNC_LDS 1
#endif

typedef float v2f __attribute__((ext_vector_type(2)));
typedef float v8f __attribute__((ext_vector_type(8)));

namespace {
constexpr int Bc = 8, Kc = 128, Nc = 2048, Dc = 32;
constexpr int TN = 8;          // n-points per block
constexpr int PAD = 32;        // vote row stride (floats); b128 LDS reads are at the
                               // 512B/wave port floor anyway, so no padding needed
constexpr int SP = 9;          // scalar-tile stride (coprime with 64 banks)
constexpr int THREADS = 256;   // 8 waves (wave32)
constexpr int NBLK = Bc * (Nc / TN);   // 2048
constexpr float kLog2Pi = 1.8378770664093453f;
constexpr float kDummy  = -4.605170185988091f;   // -2*log(10)
constexpr float kEps    = 1e-16f;

constexpr int SV_F = Kc * TN * PAD;   // 32768
constexpr int SX_F = TN * PAD;        //   256
constexpr int SP_F = Kc * SP;         //  1152 (scale / vpp / logits tiles)
constexpr int SMEM_F = SV_F + SX_F + 3 * SP_F + 32;   // 36512 floats ~143KB

// output section offsets (flattened tuple, reference return order)
constexpr size_t off_vp  = 1;
constexpr size_t off_win = off_vp  + (size_t)Bc * Kc * Nc;
constexpr size_t off_wp  = off_win + (size_t)Bc * Nc * Dc;
constexpr size_t off_sw  = off_wp  + (size_t)Bc * Nc;
constexpr size_t off_swp = off_sw  + (size_t)Bc * Nc * Dc;
constexpr size_t off_pmp = off_swp + (size_t)Bc * Nc;
constexpr size_t off_mlp = off_pmp + (size_t)Bc * Nc * (size_t)Kc;
constexpr size_t off_ml  = off_mlp + (size_t)Bc * (Kc + 1) * Nc;
constexpr size_t off_ifc = off_ml  + (size_t)Bc * (Kc + 1) * Nc;
} // namespace

// Final log_prob reduction: sum 2048 per-block partials with a ones-matrix WMMA.
// D = 1(16x4) x B(4x16) + C accumulates column sums; total = sum_n D[0,n], which is
// invariant to the (k,n) lane mapping of B, so feeding 64 partials per step is exact.
__global__ __launch_bounds__(32)
void caps_final(const float* __restrict__ part, float* __restrict__ out) {
    const int ln = threadIdx.x;
    v2f a; a.x = 1.0f; a.y = 1.0f;      // A = all-ones 16x4
    v8f c = {};
    #pragma unroll
    for (int ch = 0; ch < NBLK / 64; ++ch) {
        v2f bm;
        bm.x = part[ch * 64 + ln];
        bm.y = part[ch * 64 + 32 + ln];
        c = __builtin_amdgcn_wmma_f32_16x16x4_f32(
                false, a, false, bm, (short)0, c, false, false);
    }
    float v = (ln < 16) ? c[0] : 0.0f;  // D row 0 lives in VGPR0, lanes 0..15
    #pragma unroll
    for (int o = 16; o; o >>= 1) v += __shfl_xor(v, o, 32);
    if (ln == 0) out[0] = v * (1.0f / (float)Bc);
}

__global__ __launch_bounds__(THREADS)
void caps_main(const float* __restrict__ x, const float* __restrict__ vote,
               const float* __restrict__ scale, const float* __restrict__ vpp,
               const float* __restrict__ dummy_vote,
               float* __restrict__ out, float* __restrict__ part)
{
    extern __shared__ float smem[];
    float* sv   = smem;                 // [Kc*TN][PAD] vote tile
    float* sx   = sv + SV_F;            // [TN][PAD]    x tile
    float* ss   = sx + SX_F;            // [Kc][SP]     scale tile
    float* spp  = ss + SP_F;            // [Kc][SP]     vote_presence_prob tile
    float* spl  = spp + SP_F;           // [Kc][SP]     posterior logits -> pmp
    float* sred = spl + SP_F;           // [32]         per-n lse_post partials

    const int t   = threadIdx.x;
    const int blk = blockIdx.x;
    const int b   = blk >> 8;                 // / (Nc/TN)=256
    const int n0  = (blk & 255) * TN;

    // ---- stage x / scale / vpp tiles (regular vmem) ----
    if (t < 64) {
        const int e = t * 4, n = e >> 5, d = e & 31;
        const float4 v = *reinterpret_cast<const float4*>(
            &x[((size_t)b * Nc + (n0 + n)) * Dc + d]);
        *reinterpret_cast<float4*>(&sx[n * PAD + d]) = v;
    }
    #pragma unroll
    for (int j = 0; j < 4; ++j) {
        const int e = t + 256 * j;
        const int k = e >> 3, n = e & 7;
        const size_t gi = ((size_t)b * Kc + k) * Nc + (n0 + n);
        ss[k * SP + n]  = scale[gi];
        spp[k * SP + n] = vpp[gi];
    }

    // ---- stage vote tile: async DMA to LDS (CDNA5 path), 16B/lane/iter ----
    {
        const int kb = t >> 6;          // 0..3 : k sub-lane
        const int n  = (t >> 3) & 7;    // point within tile
        const int d  = (t & 7) * 4;     // dword4 within row
        #pragma unroll
        for (int kk = 0; kk < 32; ++kk) {
            const int k = kk * 4 + kb;
            const float* gptr =
                &vote[(((size_t)b * Kc + k) * Nc + (n0 + n)) * Dc + d];
            float* lptr = &sv[((k * TN + n) << 5) + d];
#if USE_ASYNC_LDS
            const unsigned ldsoff = (unsigned)(uintptr_t)(void*)lptr;
            const unsigned long long ga = (unsigned long long)(uintptr_t)gptr;
            asm volatile("global_load_async_to_lds_b128 %0, %1, off"
                         :: "v"(ldsoff), "v"(ga) : "memory");
#else
            *reinterpret_cast<float4*>(lptr) = *reinterpret_cast<const float4*>(gptr);
#endif
        }
    }
#if USE_ASYNC_LDS
    asm volatile("s_wait_asynccnt 0x0" ::: "memory");
#endif
    __syncthreads();

    // ---- Phase A: per-(k,n) log-likelihood, 4 pairs per thread ----
    #pragma unroll
    for (int j = 0; j < 4; ++j) {
        const int p = t + 256 * j;          // p = k*8 + n
        const int k = p >> 3, n = p & 7;
        const float* vr = &sv[p << 5];
        const float* xr = &sx[n << 5];
        float sum = 0.0f;
        #pragma unroll
        for (int i = 0; i < 8; ++i) {
            const float4 v4 = *reinterpret_cast<const float4*>(vr + i * 4);
            const float4 x4 = *reinterpret_cast<const float4*>(xr + i * 4);
            const float d0 = x4.x - v4.x, d1 = x4.y - v4.y;
            const float d2 = x4.z - v4.z, d3 = x4.w - v4.w;
            sum += d0 * d0 + d1 * d1 + d2 * d2 + d3 * d3;
        }
        const float s    = ss[k * SP + n];
        const float vlp  = -0.5f * sum / (s * s)
                           - (float)Dc * (__logf(s) + 0.5f * kLog2Pi);
        const float ml   = __logf(fmaxf(spp[k * SP + n], kEps));
        spl[k * SP + n] = ml + vlp;
        const size_t nn = (size_t)(n0 + n);
        out[off_vp + ((size_t)b * Kc + k) * Nc + nn] = (ml > kDummy) ? 1.0f : 0.0f;
        out[off_ml + ((size_t)b * (Kc + 1) + k) * Nc + nn] = ml;
    }
    __syncthreads();

    // ---- Phase A2: wave w owns point n=w; reductions over k ----
    const int ln = t & 31, n = t >> 5;
    const size_t nn = (size_t)(n0 + n);

    float plv[4], mlv[4];
    float m_pl = 2.0f * kDummy;     // dummy posterior logit
    float m_ml = kDummy;            // dummy mixing logit
    float bestv = -3.4e38f; int bestk = 0;
    #pragma unroll
    for (int j = 0; j < 4; ++j) {
        const int k = ln + 32 * j;
        const float pl = spl[k * SP + n];
        const float ml = __logf(fmaxf(spp[k * SP + n], kEps));
        plv[j] = pl; mlv[j] = ml;
        m_pl = fmaxf(m_pl, pl);
        m_ml = fmaxf(m_ml, ml);
        if (pl > bestv) { bestv = pl; bestk = k; }   // increasing k: first-max kept
    }
    #pragma unroll
    for (int o = 16; o; o >>= 1) {
        m_pl = fmaxf(m_pl, __shfl_xor(m_pl, o));
        m_ml = fmaxf(m_ml, __shfl_xor(m_ml, o));
        const float ov = __shfl_xor(bestv, o);
        const int   ok = __shfl_xor(bestk, o);
        if (ov > bestv || (ov == bestv && ok < bestk)) { bestv = ov; bestk = ok; }
    }
    float s_pl = (ln == 0) ? __expf(2.0f * kDummy - m_pl) : 0.0f;
    float s_ml = (ln == 0) ? __expf(kDummy - m_ml)        : 0.0f;
    #pragma unroll
    for (int j = 0; j < 4; ++j) {
        s_pl += __expf(plv[j] - m_pl);
        s_ml += __expf(mlv[j] - m_ml);
    }
    #pragma unroll
    for (int o = 16; o; o >>= 1) {
        s_pl += __shfl_xor(s_pl, o);
        s_ml += __shfl_xor(s_ml, o);
    }
    const float lse_post = m_pl + __logf(s_pl);
    const float lse_mix  = m_ml + __logf(s_ml);

    // pmp, posterior_mixing_prob, mixing_log_prob, soft_winner_presence
    float swp = 0.0f;
    #pragma unroll
    for (int j = 0; j < 4; ++j) {
        const int k = ln + 32 * j;
        const float pmp = __expf(plv[j] - lse_post);
        spl[k * SP + n] = pmp;                                  // reuse as pmp
        out[off_pmp + ((size_t)b * Nc + nn) * Kc + k] = pmp;
        out[off_mlp + ((size_t)b * (Kc + 1) + k) * Nc + nn] = mlv[j] - lse_mix;
        swp += pmp * spp[k * SP + n];
    }
    #pragma unroll
    for (int o = 16; o; o >>= 1) swp += __shfl_xor(swp, o);

    if (ln == 0) {
        out[off_swp + (size_t)b * Nc + nn] = swp;
        out[off_wp  + (size_t)b * Nc + nn] = spp[bestk * SP + n];
        out[off_ifc + (size_t)b * Nc + nn] = (float)bestk;
        out[off_ml  + ((size_t)b * (Kc + 1) + Kc) * Nc + nn] = kDummy;
        out[off_mlp + ((size_t)b * (Kc + 1) + Kc) * Nc + nn] = kDummy - lse_mix;
        sred[n] = lse_post;
    }
    // winner vector: lane = d
    out[off_win + ((size_t)b * Nc + nn) * Dc + ln] = sv[((bestk * TN + n) << 5) + ln];

    // ---- Phase B: soft winner einsum, vote re-read from LDS only ----
    const float pmpK = __expf(2.0f * kDummy - lse_post);
    float acc0 = pmpK * dummy_vote[nn * Dc + ln];
    float acc1 = 0.0f;
    #pragma unroll 4
    for (int k = 0; k < Kc; k += 2) {
        acc0 += spl[(k + 0) * SP + n] * sv[(((k + 0) * TN + n) << 5) + ln];
        acc1 += spl[(k + 1) * SP + n] * sv[(((k + 1) * TN + n) << 5) + ln];
    }
    out[off_sw + ((size_t)b * Nc + nn) * Dc + ln] = acc0 + acc1;

    // ---- deterministic per-block partial of log_prob ----
    __syncthreads();
    if (t == 0) {
        float p = 0.0f;
        #pragma unroll
        for (int i = 0; i < TN; ++i) p += sred[i];
        part[blk] = p;                   // every slot written every call
    }
}

extern "C" void kernel_launch(void* const* d_in, const int* in_sizes, int n_in,
                              void* d_out, int out_size, void* d_ws, size_t ws_size,
                              hipStream_t stream) {
    const float* x     = (const float*)d_in[0];
    const float* vote  = (const float*)d_in[1];
    const float* scale = (const float*)d_in[2];
    const float* vpp   = (const float*)d_in[3];
    const float* dummy = (const float*)d_in[4];
    float* out  = (float*)d_out;
    float* part = (float*)d_ws;          // NBLK floats of scratch

    caps_main<<<dim3(NBLK), dim3(THREADS), SMEM_F * sizeof(float), stream>>>(
        x, vote, scale, vpp, dummy, out, part);
    caps_final<<<1, 32, 0, stream>>>(part, out);
}